// MAGATFN_48747878810252
// MI455X (gfx1250) — compile-verified
//
#include <hip/hip_runtime.h>
#include <math.h>

#define EPSF 1e-5f
#define Bq 2
#define Tt 64
#define Nn 2048
#define Hh 128
#define NHEADS 8
#define HD 16
#define Cc 32
#define HORZ 24
#define Mrows (Bq * Nn)   /* 4096 */
#define KF (Cc * Tt)      /* 2048 */

typedef float v2f __attribute__((ext_vector_type(2)));
typedef float v8f __attribute__((ext_vector_type(8)));

// D = A(16x4,f32) * B(4x16,f32) + C(16x16,f32) -- exact fp32 matrix op on CDNA5
__device__ __forceinline__ v8f wmma_f32_16x16x4(v2f a, v2f b, v8f c) {
  return __builtin_amdgcn_wmma_f32_16x16x4_f32(false, a, false, b, (short)0, c,
                                               false, false);
}

// ---------------------------------------------------------------------------
// Kernel 1: temporal conv (1->32ch, k=3, pad 1) + BN(eval) + ReLU -> f[row][c*64+t]
// ---------------------------------------------------------------------------
__global__ void conv_feat_kernel(const float* __restrict__ x,
                                 const float* __restrict__ tcw,
                                 const float* __restrict__ tcb,
                                 const float* __restrict__ tg,
                                 const float* __restrict__ tb,
                                 float* __restrict__ f) {
  const int row = blockIdx.x;          // b*N + n
  const int b = row >> 11, n = row & (Nn - 1);
  const int t = threadIdx.x;           // 0..63
  const float* xb = x + (size_t)b * Tt * Nn + n;
  const float x0  = xb[(size_t)t * Nn];
  const float xm1 = (t > 0)      ? xb[(size_t)(t - 1) * Nn] : 0.f;
  const float xp1 = (t < Tt - 1) ? xb[(size_t)(t + 1) * Nn] : 0.f;
  const float inv = rsqrtf(1.f + EPSF);
  float* frow = f + (size_t)row * KF;
  for (int c = 0; c < Cc; ++c) {
    float v = tcw[c * 3] * xm1 + tcw[c * 3 + 1] * x0 + tcw[c * 3 + 2] * xp1 + tcb[c];
    v = v * inv * tg[c] + tb[c];
    frow[c * Tt + t] = fmaxf(v, 0.f);
  }
}

// ---------------------------------------------------------------------------
// Kernel 2: generic Y(Mx128) = epilogue( X(MxK) @ W(128xK)^T + bias )
// mode 0: bias only, mode 1: LayerNorm(gvec,bvec)+ReLU, mode 2: BN(eval)+ReLU
// 256 threads = 8 waves; wave w owns a 16x16 tile (cols w*16..), 16 rows/block.
// ---------------------------------------------------------------------------
__global__ void gemm128_kernel(const float* __restrict__ X, int K,
                               const float* __restrict__ W,
                               const float* __restrict__ bias,
                               const float* __restrict__ gvec,
                               const float* __restrict__ bvec,
                               float* __restrict__ Y, int mode) {
  __shared__ float tile[16][132];
  __shared__ float smean[16], srstd[16];
  const int tid = threadIdx.x;
  const int wave = tid >> 5, lane = tid & 31;
  const int half = lane >> 4, l16 = lane & 15;
  const int row0 = blockIdx.x * 16;
  const int col = wave * 16 + l16;

  // A frag: lanes 0-15 -> rows 0..15 K={k0,k0+1}; lanes 16-31 -> K={k0+2,k0+3}
  const float* Arow = X + (size_t)(row0 + l16) * K + half * 2;
  const float* Wcol = W + (size_t)col * K + half * 2;
  v8f acc = {};
  for (int k0 = 0; k0 < K; k0 += 4) {
    v2f a = *(const v2f*)(Arow + k0);
    v2f b = *(const v2f*)(Wcol + k0);
    acc = wmma_f32_16x16x4(a, b, acc);
  }
  const float bv = bias[col];

  if (mode == 1) {  // LayerNorm + ReLU
#pragma unroll
    for (int i = 0; i < 8; ++i) tile[i + 8 * half][col] = acc[i] + bv;
    __syncthreads();
    {
      const int r = tid >> 4, c0 = tid & 15;
      float s = 0.f, s2 = 0.f;
#pragma unroll
      for (int j = 0; j < 8; ++j) {
        float v = tile[r][c0 + j * 16];
        s += v; s2 += v * v;
      }
#pragma unroll
      for (int m = 1; m < 16; m <<= 1) {
        s  += __shfl_xor(s,  m, 16);
        s2 += __shfl_xor(s2, m, 16);
      }
      if (c0 == 0) {
        float mean = s * (1.f / 128.f);
        float var  = s2 * (1.f / 128.f) - mean * mean;
        smean[r] = mean;
        srstd[r] = rsqrtf(var + EPSF);
      }
    }
    __syncthreads();
    for (int idx = tid; idx < 16 * 128; idx += 256) {
      const int r = idx >> 7, c = idx & 127;
      float v = (tile[r][c] - smean[r]) * srstd[r] * gvec[c] + bvec[c];
      Y[(size_t)(row0 + r) * Hh + c] = fmaxf(v, 0.f);
    }
  } else if (mode == 2) {  // BN(eval) + ReLU
    const float sc = gvec[col] * rsqrtf(1.f + EPSF);
    const float bb = bvec[col];
#pragma unroll
    for (int i = 0; i < 8; ++i) {
      const int Mi = i + 8 * half;
      float v = (acc[i] + bv) * sc + bb;
      Y[(size_t)(row0 + Mi) * Hh + col] = fmaxf(v, 0.f);
    }
  } else {  // bias only
#pragma unroll
    for (int i = 0; i < 8; ++i) {
      const int Mi = i + 8 * half;
      Y[(size_t)(row0 + Mi) * Hh + col] = acc[i] + bv;
    }
  }
}

// ---------------------------------------------------------------------------
// Kernel 3: fused attention per (b, h, 16-row tile).
// Phase 1: S = QK^T/4 + adj into LDS (16x2048 strip, 128 KB of 320 KB/WGP)
// Phase 2: row softmax (16 rows x 8 lanes each, shfl_xor width-8 reductions)
// Phase 3: O = P @ V, 4 waves split the K=2048 range, LDS partial reduction.
// adj (134 MB) is streamed once per batch (Bq = 2 times); attention matrix never hits HBM.
// ---------------------------------------------------------------------------
__global__ void attn_kernel(const float* __restrict__ Q,
                            const float* __restrict__ Km,
                            const float* __restrict__ Vm,
                            const float* __restrict__ adj,
                            float* __restrict__ go) {
  extern __shared__ float S[];  // 16 * 2048 floats
  const int tid = threadIdx.x;  // 128 threads = 4 waves
  const int wave = tid >> 5, lane = tid & 31;
  const int half = lane >> 4, l16 = lane & 15;
  const int n0 = blockIdx.x * 16;
  const int h = blockIdx.y;
  const int b = blockIdx.z;

  const float* Qb = Q + (size_t)b * Nn * Hh + h * HD;
  const float* Kb = Km + (size_t)b * Nn * Hh + h * HD;
  const float* Vb = Vm + (size_t)b * Nn * Hh + h * HD;

  // Phase 1
  const float* Qrow = Qb + (size_t)(n0 + l16) * Hh + half * 2;
  for (int m0 = wave * 16; m0 < Nn; m0 += 64) {
    const float* Kcol = Kb + (size_t)(m0 + l16) * Hh + half * 2;
    v8f s = {};
#pragma unroll
    for (int k0 = 0; k0 < HD; k0 += 4) {
      v2f a = *(const v2f*)(Qrow + k0);
      v2f bb = *(const v2f*)(Kcol + k0);
      s = wmma_f32_16x16x4(a, bb, s);
    }
    const float* adjp = adj + ((size_t)h * Nn + n0) * Nn + m0 + l16;
#pragma unroll
    for (int i = 0; i < 8; ++i) {
      const int Mi = i + 8 * half;
      S[Mi * Nn + m0 + l16] = s[i] * 0.25f + adjp[(size_t)Mi * Nn];
    }
  }
  __syncthreads();

  // Phase 2: softmax over each row of 2048
  {
    const int r = tid >> 3, c0 = tid & 7;
    float* Srow = S + r * Nn;
    float mx = -3.4e38f;
    for (int j = c0; j < Nn; j += 8) mx = fmaxf(mx, Srow[j]);
#pragma unroll
    for (int m = 1; m < 8; m <<= 1) mx = fmaxf(mx, __shfl_xor(mx, m, 8));
    float sum = 0.f;
    for (int j = c0; j < Nn; j += 8) {
      float e = __expf(Srow[j] - mx);
      Srow[j] = e;
      sum += e;
    }
#pragma unroll
    for (int m = 1; m < 8; m <<= 1) sum += __shfl_xor(sum, m, 8);
    const float inv = 1.f / sum;
    for (int j = c0; j < Nn; j += 8) Srow[j] *= inv;
  }
  __syncthreads();

  // Phase 3: O = P @ V  (wave w covers m in [w*512, w*512+512))
  v8f o = {};
  for (int m0 = wave * 512; m0 < wave * 512 + 512; m0 += 4) {
    const int kk = m0 + half * 2;
    v2f a = *(const v2f*)(S + l16 * Nn + kk);   // P frag from LDS
    const float* Vp = Vb + (size_t)kk * Hh + l16;
    v2f bb;
    bb.x = Vp[0];
    bb.y = Vp[Hh];
    o = wmma_f32_16x16x4(a, bb, o);
  }
  __syncthreads();  // all P reads done; reuse LDS for partials
#pragma unroll
  for (int i = 0; i < 8; ++i)
    S[(wave * 16 + i + 8 * half) * 16 + l16] = o[i];
  __syncthreads();
  for (int idx = tid; idx < 256; idx += 128) {
    const int Mi = idx >> 4, c = idx & 15;
    float r = S[Mi * 16 + c] + S[(16 + Mi) * 16 + c] +
              S[(32 + Mi) * 16 + c] + S[(48 + Mi) * 16 + c];
    go[((size_t)b * Nn + n0 + Mi) * Hh + h * 16 + c] = r;
  }
}

// ---------------------------------------------------------------------------
// Kernel 4: s1 conv (k=2, pad 1, out len N+1) as two accumulated WMMA GEMMs
// f1[b][p] = BNReLU( gf[b][p-1] @ W0^T + gf[b][p] @ W1^T + bias ), zero-padded.
// ---------------------------------------------------------------------------
__global__ void gemm_shift_bn_kernel(const float* __restrict__ X,
                                     const float* __restrict__ Ws1, // (128,128,2)
                                     const float* __restrict__ bias,
                                     const float* __restrict__ gvec,
                                     const float* __restrict__ bvec,
                                     float* __restrict__ Y) {      // (B,2049,128)
  const int tid = threadIdx.x;
  const int wave = tid >> 5, lane = tid & 31;
  const int half = lane >> 4, l16 = lane & 15;
  const int row0 = blockIdx.x * 16;
  const int b = blockIdx.y;
  const int col = wave * 16 + l16;
  const int p = row0 + l16;
  const bool hp = (p >= 1) && (p <= Nn);
  const bool hc = (p < Nn);
  const float* Xb = X + (size_t)b * Nn * Hh;
  const float* rp = Xb + (size_t)(p - 1) * Hh + half * 2;
  const float* rc = Xb + (size_t)p * Hh + half * 2;
  const float* wb = Ws1 + (size_t)col * (Hh * 2) + half * 4;
  v8f acc = {};
  for (int k0 = 0; k0 < Hh; k0 += 4) {
    v2f ap = {0.f, 0.f}, ac = {0.f, 0.f};
    if (hp) ap = *(const v2f*)(rp + k0);
    if (hc) ac = *(const v2f*)(rc + k0);
    const float* wk = wb + k0 * 2;
    v2f b0; b0.x = wk[0]; b0.y = wk[2];
    v2f b1; b1.x = wk[1]; b1.y = wk[3];
    acc = wmma_f32_16x16x4(ap, b0, acc);
    acc = wmma_f32_16x16x4(ac, b1, acc);
  }
  const float bv = bias[col];
  const float sc = gvec[col] * rsqrtf(1.f + EPSF);
  const float bb2 = bvec[col];
#pragma unroll
  for (int i = 0; i < 8; ++i) {
    const int Mi = i + 8 * half;
    const int pr = row0 + Mi;
    if (pr <= Nn) {
      float v = (acc[i] + bv) * sc + bb2;
      Y[((size_t)b * (Nn + 1) + pr) * Hh + col] = fmaxf(v, 0.f);
    }
  }
}

// ---------------------------------------------------------------------------
// Kernel 5: linear interp of f1 (len 2049 -> 2048) + softmax-weighted fusion
// ---------------------------------------------------------------------------
__global__ void fuse_interp_kernel(const float* __restrict__ f0,
                                   const float* __restrict__ f1,
                                   const float* __restrict__ fw,
                                   float* __restrict__ fused) {
  const int idx = blockIdx.x * blockDim.x + threadIdx.x;
  if (idx >= Mrows * Hh) return;
  const int row = idx >> 7, c = idx & 127;
  const int b = row >> 11, n = row & (Nn - 1);
  const float w0 = fw[0], w1 = fw[1];
  const float mx = fmaxf(w0, w1);
  const float e0 = __expf(w0 - mx), e1 = __expf(w1 - mx);
  const float a0 = e0 / (e0 + e1), a1 = e1 / (e0 + e1);
  float src = (n + 0.5f) * ((float)(Nn + 1) / (float)Nn) - 0.5f;
  src = fminf(fmaxf(src, 0.f), (float)Nn);
  const int i0 = (int)floorf(src);
  const int i1 = min(i0 + 1, Nn);
  const float wv = src - (float)i0;
  const float* f1b = f1 + (size_t)b * (Nn + 1) * Hh + c;
  const float v1 = f1b[(size_t)i0 * Hh] * (1.f - wv) + f1b[(size_t)i1 * Hh] * wv;
  fused[idx] = a0 * f0[idx] + a1 * v1;
}

// ---------------------------------------------------------------------------
// Kernel 6: heads -- preds (24), gates (24), gated blend with x_last, transpose
// one wave per row; lane = horizon index (lanes 24..31 idle)
// ---------------------------------------------------------------------------
__global__ void head_kernel(const float* __restrict__ h2,
                            const float* __restrict__ ff,
                            const float* __restrict__ pr2w,
                            const float* __restrict__ pr2b,
                            const float* __restrict__ rgw,
                            const float* __restrict__ rgb,
                            const float* __restrict__ x,
                            float* __restrict__ out) {
  const int wave = threadIdx.x >> 5, lane = threadIdx.x & 31;
  const int row = blockIdx.x * 8 + wave;
  if (row >= Mrows || lane >= HORZ) return;
  const int b = row >> 11, n = row & (Nn - 1);
  const float* hrow = h2 + (size_t)row * Hh;
  const float* frow = ff + (size_t)row * Hh;
  const float* wp = pr2w + (size_t)lane * Hh;
  const float* wg = rgw + (size_t)lane * Hh;
  float dp = 0.f, dg = 0.f;
  for (int k2 = 0; k2 < Hh; ++k2) {
    dp += hrow[k2] * wp[k2];
    dg += frow[k2] * wg[k2];
  }
  const float pred = dp + pr2b[lane];
  const float gate = 1.f / (1.f + __expf(-(dg + rgb[lane])));
  const float xl = x[((size_t)b * Tt + (Tt - 1)) * Nn + n];
  out[((size_t)b * HORZ + lane) * Nn + n] = gate * xl + (1.f - gate) * pred;
}

// reg = 1e-4 * mean(|softmax|) = 1e-4 * (1/N) exactly (rows sum to 1, all >=0)
__global__ void reg_kernel(float* __restrict__ out) {
  if (threadIdx.x == 0 && blockIdx.x == 0)
    out[(size_t)Bq * HORZ * Nn] = 0.0001f / (float)Nn;
}

// ---------------------------------------------------------------------------
extern "C" void kernel_launch(void* const* d_in, const int* in_sizes, int n_in,
                              void* d_out, int out_size, void* d_ws,
                              size_t ws_size, hipStream_t stream) {
  (void)in_sizes; (void)n_in; (void)out_size; (void)ws_size;
  const float* x     = (const float*)d_in[0];
  const float* tc_w  = (const float*)d_in[1];
  const float* tc_b  = (const float*)d_in[2];
  const float* tc_g  = (const float*)d_in[3];
  const float* tc_bb = (const float*)d_in[4];
  const float* fp_w  = (const float*)d_in[5];
  const float* fp_b  = (const float*)d_in[6];
  const float* fp_lg = (const float*)d_in[7];
  const float* fp_lb = (const float*)d_in[8];
  const float* q_w   = (const float*)d_in[9];
  const float* q_b   = (const float*)d_in[10];
  const float* k_w   = (const float*)d_in[11];
  const float* k_b   = (const float*)d_in[12];
  const float* v_w   = (const float*)d_in[13];
  const float* v_b   = (const float*)d_in[14];
  const float* o_w   = (const float*)d_in[15];
  const float* o_b   = (const float*)d_in[16];
  const float* adj   = (const float*)d_in[17];
  const float* s0_w  = (const float*)d_in[18];
  const float* s0_b  = (const float*)d_in[19];
  const float* s0_g  = (const float*)d_in[20];
  const float* s0_bb = (const float*)d_in[21];
  const float* s1_w  = (const float*)d_in[22];
  const float* s1_b  = (const float*)d_in[23];
  const float* s1_g  = (const float*)d_in[24];
  const float* s1_bb = (const float*)d_in[25];
  const float* fwt   = (const float*)d_in[26];
  const float* fu_w  = (const float*)d_in[27];
  const float* fu_b  = (const float*)d_in[28];
  const float* fu_lg = (const float*)d_in[29];
  const float* fu_lb = (const float*)d_in[30];
  const float* pr1_w = (const float*)d_in[31];
  const float* pr1_b = (const float*)d_in[32];
  const float* pr_lg = (const float*)d_in[33];
  const float* pr_lb = (const float*)d_in[34];
  const float* pr2_w = (const float*)d_in[35];
  const float* pr2_b = (const float*)d_in[36];
  const float* rg_w  = (const float*)d_in[37];
  const float* rg_b  = (const float*)d_in[38];
  float* out = (float*)d_out;

  float* w     = (float*)d_ws;
  float* f_    = w;                                  // 4096*2048
  float* feats = f_    + (size_t)Mrows * KF;
  float* qb    = feats + (size_t)Mrows * Hh;
  float* kb    = qb    + (size_t)Mrows * Hh;
  float* vb    = kb    + (size_t)Mrows * Hh;
  float* go    = vb    + (size_t)Mrows * Hh;
  float* gf    = go    + (size_t)Mrows * Hh;
  float* f0    = gf    + (size_t)Mrows * Hh;
  float* f1    = f0    + (size_t)Mrows * Hh;         // B*(N+1)*H
  float* fused = f1    + (size_t)Bq * (Nn + 1) * Hh;
  float* ffb   = fused + (size_t)Mrows * Hh;
  float* h2    = ffb   + (size_t)Mrows * Hh;

  conv_feat_kernel<<<Mrows, Tt, 0, stream>>>(x, tc_w, tc_b, tc_g, tc_bb, f_);
  gemm128_kernel<<<Mrows / 16, 256, 0, stream>>>(f_, KF, fp_w, fp_b, fp_lg, fp_lb, feats, 1);
  gemm128_kernel<<<Mrows / 16, 256, 0, stream>>>(feats, Hh, q_w, q_b, nullptr, nullptr, qb, 0);
  gemm128_kernel<<<Mrows / 16, 256, 0, stream>>>(feats, Hh, k_w, k_b, nullptr, nullptr, kb, 0);
  gemm128_kernel<<<Mrows / 16, 256, 0, stream>>>(feats, Hh, v_w, v_b, nullptr, nullptr, vb, 0);
  attn_kernel<<<dim3(Nn / 16, NHEADS, Bq), 128, 16 * Nn * sizeof(float), stream>>>(
      qb, kb, vb, adj, go);
  gemm128_kernel<<<Mrows / 16, 256, 0, stream>>>(go, Hh, o_w, o_b, nullptr, nullptr, gf, 0);
  gemm128_kernel<<<Mrows / 16, 256, 0, stream>>>(gf, Hh, s0_w, s0_b, s0_g, s0_bb, f0, 2);
  gemm_shift_bn_kernel<<<dim3(129, Bq), 256, 0, stream>>>(gf, s1_w, s1_b, s1_g, s1_bb, f1);
  fuse_interp_kernel<<<(Mrows * Hh + 255) / 256, 256, 0, stream>>>(f0, f1, fwt, fused);
  gemm128_kernel<<<Mrows / 16, 256, 0, stream>>>(fused, Hh, fu_w, fu_b, fu_lg, fu_lb, ffb, 1);
  gemm128_kernel<<<Mrows / 16, 256, 0, stream>>>(ffb, Hh, pr1_w, pr1_b, pr_lg, pr_lb, h2, 1);
  head_kernel<<<Mrows / 8, 256, 0, stream>>>(h2, ffb, pr2_w, pr2_b, rg_w, rg_b, x, out);
  reg_kernel<<<1, 32, 0, stream>>>(out);
}